// LightGCNFull_240518168580
// MI455X (gfx1250) — compile-verified
//
#include <hip/hip_runtime.h>

typedef __attribute__((ext_vector_type(2))) float v2f;
typedef __attribute__((ext_vector_type(4))) float v4f;
typedef __attribute__((ext_vector_type(8))) float v8f;

#define B_  16
#define T_  4
#define NU_ 256
#define NP_ 512
#define NN_ 1280   // NU + 2*NP
#define F_  200
#define H_  64

// ---------------------------------------------------------------------------
// 1) Projections: rows 0..511 -> poi_p = x_poi @ fc_poi
//    rows 512..4607 -> x[b, n, :] = x_user[b,n,:] @ fc_user
// ---------------------------------------------------------------------------
__global__ void __launch_bounds__(64)
proj_kernel(const float* __restrict__ x_poi, const float* __restrict__ x_user,
            const float* __restrict__ fc_poi, const float* __restrict__ fc_user,
            float* __restrict__ poi_p, float* __restrict__ xbuf) {
  __shared__ float row[F_];
  int r = blockIdx.x;
  const float* in;
  const float* W;
  float* out;
  if (r < NP_) {
    in = x_poi + (size_t)r * F_;  W = fc_poi;  out = poi_p + (size_t)r * H_;
  } else {
    int rr = r - NP_;
    int b = rr / NU_, n = rr % NU_;
    in = x_user + ((size_t)b * NU_ + n) * F_;
    W = fc_user;
    out = xbuf + ((size_t)b * NN_ + n) * H_;
  }
  for (int k = threadIdx.x; k < F_; k += 64) row[k] = in[k];
  __syncthreads();
  int h = threadIdx.x;
  float acc = 0.f;
  for (int k = 0; k < F_; ++k) acc += row[k] * W[k * H_ + h];
  out[h] = acc;
}

// ---------------------------------------------------------------------------
// 2) Broadcast poi projection into both poi slots of x (batch-invariant)
// ---------------------------------------------------------------------------
__global__ void __launch_bounds__(256)
scatter_poi(const float* __restrict__ poi_p, float* __restrict__ xbuf) {
  int idx = blockIdx.x * 256 + threadIdx.x;        // over B*NP*H
  if (idx >= B_ * NP_ * H_) return;
  int h = idx & (H_ - 1);
  int p = (idx / H_) % NP_;
  int b = idx / (H_ * NP_);
  float val = poi_p[(size_t)p * H_ + h];
  xbuf[((size_t)b * NN_ + NU_ + p) * H_ + h] = val;
  xbuf[((size_t)b * NN_ + NU_ + NP_ + p) * H_ + h] = val;
}

// ---------------------------------------------------------------------------
// 3) A = mean_t(x_adj);  d = rsqrt(rowsum(A) + N)   (tA = A+1 row sum)
//    The one pass over the 419 MB input: B128 nontemporal loads so the
//    streamed adj does not evict the freshly written A (105 MB) from L2.
// ---------------------------------------------------------------------------
__global__ void __launch_bounds__(256)
mean_rowsum(const float* __restrict__ adj, float* __restrict__ A,
            float* __restrict__ dvec) {
  int i = blockIdx.x, b = blockIdx.y;
  float part = 0.f;
  float* Arow = A + ((size_t)b * NN_ + i) * NN_;
  const v4f* r0 = (const v4f*)(adj + (((size_t)(b * T_ + 0)) * NN_ + i) * NN_);
  const v4f* r1 = (const v4f*)(adj + (((size_t)(b * T_ + 1)) * NN_ + i) * NN_);
  const v4f* r2 = (const v4f*)(adj + (((size_t)(b * T_ + 2)) * NN_ + i) * NN_);
  const v4f* r3 = (const v4f*)(adj + (((size_t)(b * T_ + 3)) * NN_ + i) * NN_);
  for (int j4 = threadIdx.x; j4 < NN_ / 4; j4 += 256) {
    v4f a0 = __builtin_nontemporal_load(r0 + j4);
    v4f a1 = __builtin_nontemporal_load(r1 + j4);
    v4f a2 = __builtin_nontemporal_load(r2 + j4);
    v4f a3 = __builtin_nontemporal_load(r3 + j4);
    v4f s = (a0 + a1 + a2 + a3) * 0.25f;
    *(v4f*)&Arow[j4 * 4] = s;
    part += s.x + s.y + s.z + s.w;
  }
  __shared__ float red[256];
  red[threadIdx.x] = part;
  __syncthreads();
  for (int off = 128; off > 0; off >>= 1) {
    if (threadIdx.x < off) red[threadIdx.x] += red[threadIdx.x + off];
    __syncthreads();
  }
  if (threadIdx.x == 0) dvec[b * NN_ + i] = rsqrtf(red[0] + (float)NN_);
}

// ---------------------------------------------------------------------------
// 4) y = d ⊙ x (elementwise)
// ---------------------------------------------------------------------------
__global__ void __launch_bounds__(256)
make_y(const float* __restrict__ xbuf, const float* __restrict__ dvec,
       float* __restrict__ y) {
  int idx = blockIdx.x * 256 + threadIdx.x;        // over B*N*H
  if (idx >= B_ * NN_ * H_) return;
  int bn = idx / H_;
  y[idx] = dvec[bn] * xbuf[idx];
}

// ---------------------------------------------------------------------------
// 5) S[b,h] = sum_j y[b,j,h]
// ---------------------------------------------------------------------------
__global__ void __launch_bounds__(1024)
colsum(const float* __restrict__ y, float* __restrict__ S) {
  int b = blockIdx.x;
  int h = threadIdx.x, g = threadIdx.y;            // block (64, 16)
  float part = 0.f;
  for (int j = g; j < NN_; j += 16)
    part += y[((size_t)b * NN_ + j) * H_ + h];
  __shared__ float red[16][H_];
  red[g][h] = part;
  __syncthreads();
  if (g == 0) {
    float s = 0.f;
#pragma unroll
    for (int k = 0; k < 16; ++k) s += red[k][h];
    S[b * H_ + h] = s;
  }
}

// ---------------------------------------------------------------------------
// 6) WMMA f32 GEMM:  Xout[b,i,:] = d_i * ( (A[b] @ Yin[b])[i,:] + S[b,:] )
//    optionally Yout = d_i * Xout (feeds the second propagation step).
//    Block = 4 waves, 64x64 C tile; per-wave 16 M-rows x 64 N-cols.
//    Y tile is stored TRANSPOSED in LDS so each B fragment is one aligned
//    8-byte contiguous read (same as the A fragment) -> no packing movs.
//    V_WMMA_F32_16X16X4_F32 layouts per ISA 7.12.2:
//      A 16x4: lane l<16 -> {K0,K1} of row l; lane>=16 -> {K2,K3}
//      B 4x16: lane n<16 -> {K0,K1} of col n; lane>=16 -> {K2,K3}
//      C 16x16: vgpr g: lanes0-15 = M=g, lanes16-31 = M=g+8, N=lane&15
// ---------------------------------------------------------------------------
__global__ void __launch_bounds__(128)
dad_gemm(const float* __restrict__ A, const float* __restrict__ Yin,
         const float* __restrict__ dvec, const float* __restrict__ S,
         float* __restrict__ Xout, float* __restrict__ Yout, int write_y) {
  __shared__ float lA[64][68];    // [m][k], stride 68: aligned + conflict-free
  __shared__ float lYt[64][68];   // [h][k], transposed Y tile

  const int b   = blockIdx.y;
  const int m0  = blockIdx.x * 64;
  const int tid = threadIdx.x;
  const int lane = tid & 31;
  const int wave = tid >> 5;

  const float* Ab = A   + (size_t)b * NN_ * NN_;
  const float* Yb = Yin + (size_t)b * NN_ * H_;

  v8f acc[4] = {};

  const int lr    = lane & 15;
  const int khalf = (lane >> 4) << 1;              // 0 or 2

  const float* arow = &lA[wave * 16 + lr][0];
  const float* brow0 = &lYt[ 0 + lr][0];
  const float* brow1 = &lYt[16 + lr][0];
  const float* brow2 = &lYt[32 + lr][0];
  const float* brow3 = &lYt[48 + lr][0];

  for (int kt = 0; kt < NN_; kt += 64) {
    __syncthreads();
#pragma unroll
    for (int i = 0; i < 8; ++i) {                  // A tile 64x64 (M x K)
      int f4 = tid + i * 128;
      int r = f4 >> 4, c = (f4 & 15) << 2;
      v4f va = *(const v4f*)(Ab + (size_t)(m0 + r) * NN_ + kt + c);
      *(v4f*)&lA[r][c] = va;
      v4f vy = *(const v4f*)(Yb + (size_t)(kt + r) * H_ + c);
      lYt[c + 0][r] = vy.x;                        // transpose on store
      lYt[c + 1][r] = vy.y;
      lYt[c + 2][r] = vy.z;
      lYt[c + 3][r] = vy.w;
    }
    __syncthreads();

#pragma unroll
    for (int kk = 0; kk < 16; ++kk) {
      const int kb = kk * 4 + khalf;
      v2f af = *(const v2f*)(arow + kb);
      v2f b0 = *(const v2f*)(brow0 + kb);
      v2f b1 = *(const v2f*)(brow1 + kb);
      v2f b2 = *(const v2f*)(brow2 + kb);
      v2f b3 = *(const v2f*)(brow3 + kb);
      acc[0] = __builtin_amdgcn_wmma_f32_16x16x4_f32(false, af, false, b0,
                                                     (short)0, acc[0], false, false);
      acc[1] = __builtin_amdgcn_wmma_f32_16x16x4_f32(false, af, false, b1,
                                                     (short)0, acc[1], false, false);
      acc[2] = __builtin_amdgcn_wmma_f32_16x16x4_f32(false, af, false, b2,
                                                     (short)0, acc[2], false, false);
      acc[3] = __builtin_amdgcn_wmma_f32_16x16x4_f32(false, af, false, b3,
                                                     (short)0, acc[3], false, false);
    }
  }

  // epilogue: x = d_i * (acc + S[col]);  optionally y = d_i * x
  const int half = lane >> 4;
#pragma unroll
  for (int nt = 0; nt < 4; ++nt) {
    int col = nt * 16 + lr;
    float sc = S[b * H_ + col];
#pragma unroll
    for (int g = 0; g < 8; ++g) {
      int row = m0 + wave * 16 + half * 8 + g;
      float dv = dvec[b * NN_ + row];
      float xv = dv * (acc[nt][g] + sc);
      size_t o = ((size_t)b * NN_ + row) * H_ + col;
      Xout[o] = xv;
      if (write_y) Yout[o] = dv * xv;
    }
  }
}

// ---------------------------------------------------------------------------
// 7) x3 = (x + x1 + x2) / 3
// ---------------------------------------------------------------------------
__global__ void __launch_bounds__(256)
x3_kernel(const float* __restrict__ x, const float* __restrict__ x1,
          const float* __restrict__ x2, float* __restrict__ x3) {
  int i = blockIdx.x * 256 + threadIdx.x;
  if (i >= B_ * NN_ * H_) return;
  x3[i] = (x[i] + x1[i] + x2[i]) * (1.0f / 3.0f);
}

// ---------------------------------------------------------------------------
// 8) score[b,u,v] = <x3[b,u,:], x3[b,NU+v,:]>   (16x16 output tile / block)
// ---------------------------------------------------------------------------
__global__ void __launch_bounds__(256)
score_kernel(const float* __restrict__ x3, float* __restrict__ out) {
  __shared__ float xu[16][H_ + 1];   // +1 pad kills stride-64 bank conflicts
  __shared__ float xp[16][H_ + 1];
  int b  = blockIdx.z;
  int u0 = blockIdx.y * 16;
  int v0 = blockIdx.x * 16;
  int tid = threadIdx.y * 16 + threadIdx.x;
  const float* xb = x3 + (size_t)b * NN_ * H_;
#pragma unroll
  for (int i = 0; i < 4; ++i) {
    int e = tid + i * 256;
    int r = e >> 6, c = e & 63;
    xu[r][c] = xb[(size_t)(u0 + r) * H_ + c];
    xp[r][c] = xb[(size_t)(NU_ + v0 + r) * H_ + c];
  }
  __syncthreads();
  float acc = 0.f;
#pragma unroll
  for (int h = 0; h < H_; ++h)
    acc += xu[threadIdx.y][h] * xp[threadIdx.x][h];
  out[((size_t)b * NU_ + u0 + threadIdx.y) * NP_ + v0 + threadIdx.x] = acc;
}

// ---------------------------------------------------------------------------
extern "C" void kernel_launch(void* const* d_in, const int* in_sizes, int n_in,
                              void* d_out, int out_size, void* d_ws, size_t ws_size,
                              hipStream_t stream) {
  (void)in_sizes; (void)n_in; (void)out_size; (void)ws_size;
  const float* x_poi   = (const float*)d_in[0];
  const float* x_user  = (const float*)d_in[1];
  const float* x_adj   = (const float*)d_in[2];
  const float* fc_poi  = (const float*)d_in[3];
  const float* fc_user = (const float*)d_in[4];

  float* ws = (float*)d_ws;
  size_t off = 0;
  float* A     = ws + off; off += (size_t)B_ * NN_ * NN_;  // 26,214,400
  float* xb    = ws + off; off += (size_t)B_ * NN_ * H_;   // 1,310,720
  float* y     = ws + off; off += (size_t)B_ * NN_ * H_;   // also reused as x3
  float* y1    = ws + off; off += (size_t)B_ * NN_ * H_;
  float* x1    = ws + off; off += (size_t)B_ * NN_ * H_;
  float* x2    = ws + off; off += (size_t)B_ * NN_ * H_;
  float* dvec  = ws + off; off += (size_t)B_ * NN_;
  float* S     = ws + off; off += (size_t)B_ * H_;
  float* S1    = ws + off; off += (size_t)B_ * H_;
  float* poi_p = ws + off; off += (size_t)NP_ * H_;

  // 1-2) feature projections -> xb (B, N, H)
  proj_kernel<<<NP_ + B_ * NU_, 64, 0, stream>>>(x_poi, x_user, fc_poi, fc_user,
                                                 poi_p, xb);
  scatter_poi<<<(B_ * NP_ * H_ + 255) / 256, 256, 0, stream>>>(poi_p, xb);

  // 3) A = mean_t(adj), d = rsqrt(rowsum + N)
  mean_rowsum<<<dim3(NN_, B_), 256, 0, stream>>>(x_adj, A, dvec);

  // 4-5) y = d*x, S = colsum(y)
  make_y<<<(B_ * NN_ * H_ + 255) / 256, 256, 0, stream>>>(xb, dvec, y);
  colsum<<<B_, dim3(64, 16), 0, stream>>>(y, S);

  // 6) x1 = d*(A@y + S), y1 = d*x1   [WMMA]
  dad_gemm<<<dim3(NN_ / 64, B_), 128, 0, stream>>>(A, y, dvec, S, x1, y1, 1);

  // second propagation: S1 = colsum(y1), x2 = d*(A@y1 + S1)   [WMMA]
  colsum<<<B_, dim3(64, 16), 0, stream>>>(y1, S1);
  dad_gemm<<<dim3(NN_ / 64, B_), 128, 0, stream>>>(A, y1, dvec, S1, x2, nullptr, 0);

  // 7) x3 = (x + x1 + x2)/3  (into y buffer, dead by now)
  float* x3 = y;
  x3_kernel<<<(B_ * NN_ * H_ + 255) / 256, 256, 0, stream>>>(xb, x1, x2, x3);

  // 8) score = x3_users @ x3_poi^T -> (B,1,256,512)
  score_kernel<<<dim3(NP_ / 16, NU_ / 16, B_), dim3(16, 16), 0, stream>>>(
      x3, (float*)d_out);
}